// LKA_dynamic_51110110822619
// MI455X (gfx1250) — compile-verified
//
#include <hip/hip_runtime.h>

#define BB 8
#define CC 64
#define HH 256
#define WW 256
#define KK 3
#define PLANE (HH * WW) // 65536

typedef __attribute__((ext_vector_type(2))) float v2f;
typedef __attribute__((ext_vector_type(8))) float v8f;

__device__ __forceinline__ float gelu_exact(float v) {
    // exact GELU: 0.5*x*(1+erf(x/sqrt(2)))
    return 0.5f * v * (1.0f + erff(v * 0.70710678118654752f));
}

// ---------------- adaptive avg pool to 1x1 (float4-wide, one block/plane) ----
__global__ void pool_mean_kernel(const float* __restrict__ in,
                                 float* __restrict__ pooled) {
    __shared__ float red[256];
    const int plane = blockIdx.x; // b*CC + c
    const float4* ip = (const float4*)(in + (size_t)plane * PLANE);
    float s = 0.f;
    for (int i = threadIdx.x; i < PLANE / 4; i += 256) {
        const float4 v = ip[i];
        s += (v.x + v.y) + (v.z + v.w);
    }
    red[threadIdx.x] = s;
    __syncthreads();
    for (int off = 128; off > 0; off >>= 1) {
        if (threadIdx.x < off) red[threadIdx.x] += red[threadIdx.x + off];
        __syncthreads();
    }
    if (threadIdx.x == 0) pooled[plane] = red[0] * (1.0f / PLANE);
}

// -------- reduce per-row partial sums (written by dwconv1) into pooled ------
__global__ void reduce_partials_kernel(const float* __restrict__ partials,
                                       float* __restrict__ pooled) {
    __shared__ float red[256];
    const int plane = blockIdx.x; // b*CC + c
    red[threadIdx.x] = partials[plane * HH + threadIdx.x];
    __syncthreads();
    for (int off = 128; off > 0; off >>= 1) {
        if (threadIdx.x < off) red[threadIdx.x] += red[threadIdx.x + off];
        __syncthreads();
    }
    if (threadIdx.x == 0) pooled[plane] = red[0] * (1.0f / PLANE);
}

// ------------- routing (sigmoid of pooled @ rw^T + rb) + expert kernel mix ---
template <int KS2>
__global__ void route_mix_kernel(const float* __restrict__ pooled,
                                 const float* __restrict__ rw, // [K,C]
                                 const float* __restrict__ rb, // [K]
                                 const float* __restrict__ w,  // [K,C,1,ks,ks]
                                 float* __restrict__ wk) {     // [B,C,ks*ks]
    __shared__ float r[BB * KK];
    const int tid = threadIdx.x;
    if (tid < BB * KK) {
        const int b = tid / KK, k = tid % KK;
        float z = rb[k];
        for (int c = 0; c < CC; ++c) z += pooled[b * CC + c] * rw[k * CC + c];
        r[tid] = 1.0f / (1.0f + expf(-z));
    }
    __syncthreads();
    const int total = BB * CC * KS2;
    for (int i = tid; i < total; i += 256) {
        const int b   = i / (CC * KS2);
        const int rem = i - b * (CC * KS2);
        const int c   = rem / KS2;
        const int t   = rem - c * KS2;
        float s = 0.f;
#pragma unroll
        for (int k = 0; k < KK; ++k) s += r[b * KK + k] * w[(k * CC + c) * KS2 + t];
        wk[i] = s;
    }
}

// ------------- per-sample depthwise conv (mixed kernels) + bias + GELU -------
// Optionally emits one partial sum of its 256 outputs per block (fused pooling
// for the NEXT CondConv's routing -> saves a full 134MB tensor re-read).
template <int KS, int PAD, int DIL, bool EMIT_PARTIAL>
__global__ void dwconv_gelu_kernel(const float* __restrict__ in,
                                   const float* __restrict__ wk,   // [B,C,KS*KS]
                                   const float* __restrict__ bias, // [C]
                                   float* __restrict__ out,
                                   float* __restrict__ partials) { // [B*C*HH]
    __shared__ float red[256];
    const int h  = blockIdx.x & (HH - 1);
    const int bc = blockIdx.x >> 8; // b*CC + c
    const int c  = bc & (CC - 1);
    const int w  = threadIdx.x;
    const float* ip = in + (size_t)bc * PLANE;
    const float* kp = wk + bc * KS * KS;
    float kreg[KS * KS];
#pragma unroll
    for (int t = 0; t < KS * KS; ++t) kreg[t] = kp[t];
    float s = 0.f;
#pragma unroll
    for (int dy = 0; dy < KS; ++dy) {
        const int yy = h + dy * DIL - PAD;
        if (yy < 0 || yy >= HH) continue;
#pragma unroll
        for (int dx = 0; dx < KS; ++dx) {
            const int xx = w + dx * DIL - PAD;
            if (xx >= 0 && xx < WW) s += ip[yy * WW + xx] * kreg[dy * KS + dx];
        }
    }
    const float g = gelu_exact(s + bias[c]);
    out[(size_t)bc * PLANE + h * WW + w] = g;
    if (EMIT_PARTIAL) {
        red[threadIdx.x] = g;
        __syncthreads();
        for (int off = 128; off > 0; off >>= 1) {
            if (threadIdx.x < off) red[threadIdx.x] += red[threadIdx.x + off];
            __syncthreads();
        }
        if (threadIdx.x == 0) partials[blockIdx.x] = red[0];
    }
}

// ------------- 1x1 conv (C=64 -> C=64 GEMM) via WMMA f32 + bias + x*attn ----
// Each wave: 32 contiguous pixels (2 N-tiles sharing A-fragments) x 64 out-ch.
//   A (16x4) = wp tile   (M = out channel, K = in channel)  -- from LDS
//   B (4x16) = attn tile (K = in channel, N = pixel)        -- from global
//   D (16x16) rows = out channels, cols = contiguous pixels -> coalesced store
__global__ void proj_mul_wmma_kernel(const float* __restrict__ attn,
                                     const float* __restrict__ wp, // [O,C]
                                     const float* __restrict__ bp, // [O]
                                     const float* __restrict__ x,
                                     float* __restrict__ out) {
    __shared__ float swp[CC * CC]; // 16 KB
    for (int i = threadIdx.x; i < CC * CC; i += 256) swp[i] = wp[i];
    __syncthreads();

    const int lane = threadIdx.x & 31;
    const int wave = threadIdx.x >> 5;
    const int l16  = lane & 15;
    const int h16  = lane >> 4;

    const int tile = blockIdx.x * 8 + wave; // each tile = 32 pixels
    const int p0   = tile * 32;
    const int b    = p0 >> 16;          // plane = 65536 pixels
    const int p    = p0 & (PLANE - 1);  // pixel offset within plane

    const float* ab = attn + (size_t)b * CC * PLANE + p + l16;

    v8f acc[2][4] = {};

#pragma unroll
    for (int ks = 0; ks < 16; ++ks) {
        const int c0 = ks * 4;
        // B fragments: lane holds (K = h16, N) and (K = 2+h16, N)
        v2f bf0, bf1;
        bf0.x = ab[(size_t)(c0 + h16) * PLANE];
        bf0.y = ab[(size_t)(c0 + 2 + h16) * PLANE];
        bf1.x = ab[(size_t)(c0 + h16) * PLANE + 16];
        bf1.y = ab[(size_t)(c0 + 2 + h16) * PLANE + 16];
#pragma unroll
        for (int ot = 0; ot < 4; ++ot) {
            const int o = ot * 16 + l16; // A row (out channel)
            // A fragment: lane holds (M = l16, K = 2*h16) and (M = l16, K = 2*h16+1)
            const v2f afrag = *(const v2f*)(&swp[o * CC + c0 + 2 * h16]);
            acc[0][ot] = __builtin_amdgcn_wmma_f32_16x16x4_f32(
                false, afrag, false, bf0, (short)0, acc[0][ot], false, false);
            acc[1][ot] = __builtin_amdgcn_wmma_f32_16x16x4_f32(
                false, afrag, false, bf1, (short)0, acc[1][ot], false, false);
        }
    }

    const size_t base = (size_t)b * CC * PLANE + p + l16;
#pragma unroll
    for (int nt = 0; nt < 2; ++nt) {
#pragma unroll
        for (int ot = 0; ot < 4; ++ot) {
#pragma unroll
            for (int rr = 0; rr < 8; ++rr) {
                const int o = ot * 16 + 8 * h16 + rr; // D row M = rr (+8 hi half)
                const size_t idx = base + nt * 16 + (size_t)o * PLANE;
                const float v = acc[nt][ot][rr] + bp[o];
                out[idx] = x[idx] * v;
            }
        }
    }
}

extern "C" void kernel_launch(void* const* d_in, const int* in_sizes, int n_in,
                              void* d_out, int out_size, void* d_ws, size_t ws_size,
                              hipStream_t stream) {
    const float* x   = (const float*)d_in[0];
    const float* w0  = (const float*)d_in[1];
    const float* b0  = (const float*)d_in[2];
    const float* r0w = (const float*)d_in[3];
    const float* r0b = (const float*)d_in[4];
    const float* w1  = (const float*)d_in[5];
    const float* b1  = (const float*)d_in[6];
    const float* r1w = (const float*)d_in[7];
    const float* r1b = (const float*)d_in[8];
    const float* wp  = (const float*)d_in[9];
    const float* bp  = (const float*)d_in[10];
    float* out = (float*)d_out;

    // ws layout: attn2 (B*C*H*W) | pooled (B*C) | wk0 | wk1 | row partials
    float* ws       = (float*)d_ws;
    float* attn2    = ws;
    float* pooled   = ws + (size_t)BB * CC * PLANE;
    float* wk0      = pooled + BB * CC;
    float* wk1      = wk0 + BB * CC * 25;
    float* partials = wk1 + BB * CC * 49; // BB*CC*HH floats (512 KB)

    // Stage 1: CondConv depthwise 5x5 (attn1 -> d_out), fused row-sum partials
    pool_mean_kernel<<<BB * CC, 256, 0, stream>>>(x, pooled);
    route_mix_kernel<25><<<1, 256, 0, stream>>>(pooled, r0w, r0b, w0, wk0);
    dwconv_gelu_kernel<5, 2, 1, true>
        <<<BB * CC * HH, 256, 0, stream>>>(x, wk0, b0, out, partials);

    // Stage 2: routing from fused partials (no full-tensor re-read), then
    // CondConv depthwise 7x7, pad 9, dil 3 (attn2 -> ws)
    reduce_partials_kernel<<<BB * CC, 256, 0, stream>>>(partials, pooled);
    route_mix_kernel<49><<<1, 256, 0, stream>>>(pooled, r1w, r1b, w1, wk1);
    dwconv_gelu_kernel<7, 9, 3, false>
        <<<BB * CC * HH, 256, 0, stream>>>(out, wk1, b1, attn2, nullptr);

    // Stage 3: 1x1 projection via WMMA f32 + bias, fused with out = x * attn
    proj_mul_wmma_kernel<<<(BB * PLANE) / 256, 256, 0, stream>>>(attn2, wp, bp, x, out);
}